// GAT_LP_78391743086867
// MI455X (gfx1250) — compile-verified
//
#include <hip/hip_runtime.h>
#include <hip/hip_bf16.h>
#include <math.h>

typedef float v2f __attribute__((ext_vector_type(2)));
typedef float v8f __attribute__((ext_vector_type(8)));

#define NMN 8192
#define NDN 6144
#define NTN 14336
#define EMN 131072
#define EDN 98304
#define EGN 229376

// ---------------------------------------------------------------- utilities

__global__ void fill_f32(float* __restrict__ p, float v, long n) {
  long t = (long)blockIdx.x * blockDim.x + threadIdx.x;
  if (t < n) p[t] = v;
}

__global__ void gather_ew(const float* __restrict__ mat, const int* __restrict__ src,
                          const int* __restrict__ dst, float* __restrict__ ew,
                          int E, int ld) {
  int e = blockIdx.x * blockDim.x + threadIdx.x;
  if (e < E) ew[e] = mat[(long)src[e] * ld + dst[e]];
}

__global__ void deg_scatter(const float* __restrict__ ew, const int* __restrict__ dst,
                            float* __restrict__ deg, int E) {
  int e = blockIdx.x * blockDim.x + threadIdx.x;
  if (e < E) atomicAdd(&deg[dst[e]], ew[e]);
}

__global__ void deg_to_dinv(float* __restrict__ deg, int n) {
  int i = blockIdx.x * blockDim.x + threadIdx.x;
  if (i < n) {
    float v = deg[i];
    deg[i] = (v > 0.f) ? rsqrtf(v) : 0.f;
  }
}

// ---------------------------------------------------------------- WMMA GEMM
// Y[n,F] = X[n,K] @ W[K,F] (+bias)(+relu). One wave per 16x64 tile.
// f32 WMMA 16x16x4: A lanes 0-15 hold (K=0,1), lanes 16-31 hold (K=2,3);
// B symmetric; C/D: vgpr v, lanes 0-15 -> M=v, lanes 16-31 -> M=v+8.
__global__ void wmma_gemm_f32(const float* __restrict__ X, const float* __restrict__ W,
                              float* __restrict__ Y, const float* __restrict__ bias,
                              int n, int K, int F, int relu) {
  const int tileM = blockIdx.x;
  const int tileN = blockIdx.y;
  const int lane  = threadIdx.x;           // 0..31
  const int mrow  = tileM * 16 + (lane & 15);
  const int khalf = (lane >> 4) * 2;
  const int ncol0 = tileN * 64 + (lane & 15);

  v8f acc[4];
#pragma unroll
  for (int t = 0; t < 4; ++t) acc[t] = (v8f){0.f,0.f,0.f,0.f,0.f,0.f,0.f,0.f};

  const float* __restrict__ Xr = X + (long)mrow * K;
  for (int k = 0; k < K; k += 4) {
    v2f a;
    a.x = Xr[k + khalf];
    a.y = Xr[k + khalf + 1];
    const float* __restrict__ w0 = W + (long)(k + khalf) * F + ncol0;
    const float* __restrict__ w1 = w0 + F;
#pragma unroll
    for (int t = 0; t < 4; ++t) {
      v2f b;
      b.x = w0[16 * t];
      b.y = w1[16 * t];
      acc[t] = __builtin_amdgcn_wmma_f32_16x16x4_f32(
          false, a, false, b, (short)0, acc[t], false, false);
    }
  }

  const int rhalf = (lane >> 4) * 8;
#pragma unroll
  for (int v = 0; v < 8; ++v) {
    long r = (long)tileM * 16 + rhalf + v;
#pragma unroll
    for (int t = 0; t < 4; ++t) {
      int col = ncol0 + 16 * t;
      float val = acc[t][v];
      if (bias) val += bias[col];
      if (relu) val = fmaxf(val, 0.f);
      Y[r * F + col] = val;
    }
  }
}

// ---------------------------------------------------------------- GCN graph ops
// wave per edge, lane covers 4 of 128 feature channels (float4).
__global__ void gcn_scatter(const float* __restrict__ h, const int* __restrict__ src,
                            const int* __restrict__ dst, const float* __restrict__ ew,
                            const float* __restrict__ dinv, float* __restrict__ acc, int E) {
  int t = blockIdx.x * blockDim.x + threadIdx.x;
  int e = t >> 5, lane = t & 31;
  if (e >= E) return;
  int s = src[e], d = dst[e];
  float norm = dinv[s] * ew[e] * dinv[d];
  const float4* hs = (const float4*)(h + (long)s * 128);
  float* out = acc + (long)d * 128 + lane * 4;
  float4 v = hs[lane];
  atomicAdd(out + 0, v.x * norm);
  atomicAdd(out + 1, v.y * norm);
  atomicAdd(out + 2, v.z * norm);
  atomicAdd(out + 3, v.w * norm);
}

// out = relu?( acc + h*dinv^2 (self loop) + bias )
__global__ void gcn_epilogue(const float* __restrict__ acc, const float* __restrict__ h,
                             const float* __restrict__ dinv, const float* __restrict__ bias,
                             float* __restrict__ out, int n, int relu) {
  long t = (long)blockIdx.x * blockDim.x + threadIdx.x;
  if (t >= (long)n * 128) return;
  int i = (int)(t >> 7), f = (int)(t & 127);
  float di = dinv[i];
  float v = acc[t] + h[t] * di * di + bias[f];
  out[t] = relu ? fmaxf(v, 0.f) : v;
}

// ---------------------------------------------------------------- GATv2 graph ops

__device__ __forceinline__ void atomicMaxFloat(float* addr, float val) {
  if (val >= 0.f) atomicMax((int*)addr, __float_as_int(val));
  else            atomicMin((unsigned int*)addr, __float_as_uint(val));
}

// wave per edge: logits[e,h] = sum_c att[h,c]*lrelu(xl[s,h,c]+xr[d,h,c]); atomic segment-max.
__global__ void gat_logits(const float* __restrict__ xl, const float* __restrict__ xr,
                           const int* __restrict__ src, const int* __restrict__ dst,
                           const float* __restrict__ att, float* __restrict__ logits,
                           float* __restrict__ mx, int E, int Etot, int C) {
  int t = blockIdx.x * blockDim.x + threadIdx.x;
  int e = t >> 5, lane = t & 31;
  if (e >= Etot) return;
  int s, d;
  if (e < E) { s = src[e]; d = dst[e]; } else { s = d = e - E; }
  const float* pl = xl + (long)s * 4 * C;
  const float* pr = xr + (long)d * 4 * C;
#pragma unroll
  for (int h = 0; h < 4; ++h) {
    float sum = 0.f;
    for (int c = lane; c < C; c += 32) {
      float v = pl[h * C + c] + pr[h * C + c];
      v = (v > 0.f) ? v : 0.2f * v;
      sum += v * att[h * C + c];
    }
#pragma unroll
    for (int off = 16; off > 0; off >>= 1) sum += __shfl_xor(sum, off, 32);
    if (lane == 0) {
      logits[(long)e * 4 + h] = sum;
      atomicMaxFloat(&mx[(long)d * 4 + h], sum);
    }
  }
}

// thread per (e,h): ex = exp(logit - mx[dst]); denom atomic-add; store ex in place.
__global__ void gat_exp(const int* __restrict__ dst, float* __restrict__ logits,
                        const float* __restrict__ mx, float* __restrict__ denom,
                        int E, int Etot) {
  long t = (long)blockIdx.x * blockDim.x + threadIdx.x;
  if (t >= (long)Etot * 4) return;
  int e = (int)(t >> 2), h = (int)(t & 3);
  int d = (e < E) ? dst[e] : e - E;
  float ex = __expf(logits[t] - mx[(long)d * 4 + h]);
  logits[t] = ex;
  atomicAdd(&denom[(long)d * 4 + h], ex);
}

// wave per edge: acc[d,h,c] += xl[s,h,c] * ex/denom
__global__ void gat_scatter(const float* __restrict__ xl, const int* __restrict__ src,
                            const int* __restrict__ dst, const float* __restrict__ exv,
                            const float* __restrict__ denom, float* __restrict__ acc,
                            int E, int Etot, int C, int logC) {
  int t = blockIdx.x * blockDim.x + threadIdx.x;
  int e = t >> 5, lane = t & 31;
  if (e >= Etot) return;
  int s, d;
  if (e < E) { s = src[e]; d = dst[e]; } else { s = d = e - E; }
  int HC = 4 * C;
  const float* pl = xl + (long)s * HC;
  float* pa = acc + (long)d * HC;
  for (int idx = lane; idx < HC; idx += 32) {
    int h = idx >> logC;
    float alpha = exv[(long)e * 4 + h] / denom[(long)d * 4 + h];
    atomicAdd(&pa[idx], pl[idx] * alpha);
  }
}

// out[i,c] = (elu?)( mean_h acc[i,h,c] + bias[c] )
__global__ void gat_finalize(const float* __restrict__ acc, const float* __restrict__ bias,
                             float* __restrict__ out, int n, int C, int elu) {
  long t = (long)blockIdx.x * blockDim.x + threadIdx.x;
  if (t >= (long)n * C) return;
  int i = (int)(t / C), c = (int)(t % C);
  const float* p = acc + (long)i * 4 * C + c;
  float s = (p[0] + p[C] + p[2 * C] + p[3 * C]) * 0.25f + bias[c];
  if (elu) s = (s > 0.f) ? s : (__expf(s) - 1.f);
  out[t] = s;
}

// xjk[i] = [x0(128) | x1(64) | x2(32)]
__global__ void concat_jk(const float* __restrict__ x0, const float* __restrict__ x1,
                          const float* __restrict__ x2, float* __restrict__ xjk, int n) {
  long t = (long)blockIdx.x * blockDim.x + threadIdx.x;
  if (t >= (long)n * 224) return;
  int i = (int)(t / 224), c = (int)(t % 224);
  float v;
  if (c < 128)      v = x0[(long)i * 128 + c];
  else if (c < 192) v = x1[(long)i * 64 + (c - 128)];
  else              v = x2[(long)i * 32 + (c - 192)];
  xjk[(long)i * 224 + c] = v;
}

// ---------------------------------------------------------------- host side

static inline int cdiv(long a, int b) { return (int)((a + b - 1) / b); }

static void launch_gemm(const float* X, const float* W, float* Y, const float* bias,
                        int n, int K, int F, int relu, hipStream_t s) {
  dim3 grid(n / 16, F / 64), block(32);
  wmma_gemm_f32<<<grid, block, 0, s>>>(X, W, Y, bias, n, K, F, relu);
}

extern "C" void kernel_launch(void* const* d_in, const int* in_sizes, int n_in,
                              void* d_out, int out_size, void* d_ws, size_t ws_size,
                              hipStream_t stream) {
  const float* mm_f  = (const float*)d_in[0];
  const float* d_s   = (const float*)d_in[1];
  const float* msmat = (const float*)d_in[2];
  const float* dsmat = (const float*)d_in[3];
  const int* msrc = (const int*)d_in[4];  const int* mdst = msrc + EMN;
  const int* dsrc = (const int*)d_in[5];  const int* ddst = dsrc + EDN;
  const int* gsrc = (const int*)d_in[6];  const int* gdst = gsrc + EGN;
  const float* W_m1 = (const float*)d_in[7];  const float* b_m1 = (const float*)d_in[8];
  const float* W_m2 = (const float*)d_in[9];  const float* b_m2 = (const float*)d_in[10];
  const float* W_d1 = (const float*)d_in[11]; const float* b_d1 = (const float*)d_in[12];
  const float* W_d2 = (const float*)d_in[13]; const float* b_d2 = (const float*)d_in[14];
  const float* Wl1  = (const float*)d_in[15]; const float* Wr1  = (const float*)d_in[16];
  const float* att1 = (const float*)d_in[17]; const float* bias1= (const float*)d_in[18];
  const float* Wl2  = (const float*)d_in[19]; const float* Wr2  = (const float*)d_in[20];
  const float* att2 = (const float*)d_in[21]; const float* bias2= (const float*)d_in[22];
  const float* W_jk = (const float*)d_in[23]; const float* b_jk = (const float*)d_in[24];

  // ---- workspace layout (floats) ----
  float* ws = (float*)d_ws;
  float* x0 = ws;                                    // NTN*128
  float* x1 = x0 + (long)NTN * 128;                  // NTN*64
  float* x2 = x1 + (long)NTN * 64;                   // NTN*32
  float* T  = x2 + (long)NTN * 32;                   // transient arena
  // GCN-phase views of T (sized for the larger m-graph)
  float* ew   = T;
  float* dinv = ew + EMN;
  float* hA   = dinv + NMN;
  float* hB   = hA + (long)NMN * 128;
  // GAT-phase views of T
  float* xl   = T;
  float* xr   = xl + (long)NTN * 256;
  float* accg = xr + (long)NTN * 256;
  float* lg   = accg + (long)NTN * 256;              // (EGN+NTN)*4
  float* mx   = lg + (long)(EGN + NTN) * 4;
  float* dn   = mx + (long)NTN * 4;
  float* xjk  = T;                                   // NTN*224 (after GAT)

  const int B = 256;
  const int Etot = EGN + NTN;

  // ================= GCN on m-graph =================
  gather_ew<<<cdiv(EMN, B), B, 0, stream>>>(msmat, msrc, mdst, ew, EMN, NMN);
  fill_f32<<<cdiv(NMN, B), B, 0, stream>>>(dinv, 1.0f, NMN);          // self-loop weight
  deg_scatter<<<cdiv(EMN, B), B, 0, stream>>>(ew, mdst, dinv, EMN);
  deg_to_dinv<<<cdiv(NMN, B), B, 0, stream>>>(dinv, NMN);
  // layer 1: m1 = relu(seg_sum + b)
  launch_gemm(mm_f, W_m1, hA, nullptr, NMN, 128, 128, 0, stream);
  fill_f32<<<cdiv((long)NMN * 128, B), B, 0, stream>>>(hB, 0.f, (long)NMN * 128);
  gcn_scatter<<<cdiv((long)EMN * 32, B), B, 0, stream>>>(hA, msrc, mdst, ew, dinv, hB, EMN);
  gcn_epilogue<<<cdiv((long)NMN * 128, B), B, 0, stream>>>(hB, hA, dinv, b_m1, hB, NMN, 1);
  // layer 2: x0[:NMN] = relu(m2)
  launch_gemm(hB, W_m2, hA, nullptr, NMN, 128, 128, 0, stream);
  fill_f32<<<cdiv((long)NMN * 128, B), B, 0, stream>>>(x0, 0.f, (long)NMN * 128);
  gcn_scatter<<<cdiv((long)EMN * 32, B), B, 0, stream>>>(hA, msrc, mdst, ew, dinv, x0, EMN);
  gcn_epilogue<<<cdiv((long)NMN * 128, B), B, 0, stream>>>(x0, hA, dinv, b_m2, x0, NMN, 1);

  // ================= GCN on d-graph =================
  float* x0d = x0 + (long)NMN * 128;
  gather_ew<<<cdiv(EDN, B), B, 0, stream>>>(dsmat, dsrc, ddst, ew, EDN, NDN);
  fill_f32<<<cdiv(NDN, B), B, 0, stream>>>(dinv, 1.0f, NDN);
  deg_scatter<<<cdiv(EDN, B), B, 0, stream>>>(ew, ddst, dinv, EDN);
  deg_to_dinv<<<cdiv(NDN, B), B, 0, stream>>>(dinv, NDN);
  launch_gemm(d_s, W_d1, hA, nullptr, NDN, 128, 128, 0, stream);
  fill_f32<<<cdiv((long)NDN * 128, B), B, 0, stream>>>(hB, 0.f, (long)NDN * 128);
  gcn_scatter<<<cdiv((long)EDN * 32, B), B, 0, stream>>>(hA, dsrc, ddst, ew, dinv, hB, EDN);
  gcn_epilogue<<<cdiv((long)NDN * 128, B), B, 0, stream>>>(hB, hA, dinv, b_d1, hB, NDN, 1);
  launch_gemm(hB, W_d2, hA, nullptr, NDN, 128, 128, 0, stream);
  fill_f32<<<cdiv((long)NDN * 128, B), B, 0, stream>>>(x0d, 0.f, (long)NDN * 128);
  gcn_scatter<<<cdiv((long)EDN * 32, B), B, 0, stream>>>(hA, dsrc, ddst, ew, dinv, x0d, EDN);
  gcn_epilogue<<<cdiv((long)NDN * 128, B), B, 0, stream>>>(x0d, hA, dinv, b_d2, x0d, NDN, 1);

  // ================= GATv2 layer 1 (C=64) =================
  launch_gemm(x0, Wl1, xl, nullptr, NTN, 128, 256, 0, stream);
  launch_gemm(x0, Wr1, xr, nullptr, NTN, 128, 256, 0, stream);
  fill_f32<<<cdiv((long)NTN * 4, B), B, 0, stream>>>(mx, -INFINITY, (long)NTN * 4);
  fill_f32<<<cdiv((long)NTN * 4, B), B, 0, stream>>>(dn, 0.f, (long)NTN * 4);
  fill_f32<<<cdiv((long)NTN * 256, B), B, 0, stream>>>(accg, 0.f, (long)NTN * 256);
  gat_logits<<<cdiv((long)Etot * 32, B), B, 0, stream>>>(xl, xr, gsrc, gdst, att1, lg, mx, EGN, Etot, 64);
  gat_exp<<<cdiv((long)Etot * 4, B), B, 0, stream>>>(gdst, lg, mx, dn, EGN, Etot);
  gat_scatter<<<cdiv((long)Etot * 32, B), B, 0, stream>>>(xl, gsrc, gdst, lg, dn, accg, EGN, Etot, 64, 6);
  gat_finalize<<<cdiv((long)NTN * 64, B), B, 0, stream>>>(accg, bias1, x1, NTN, 64, 1);

  // ================= GATv2 layer 2 (C=32) =================
  launch_gemm(x1, Wl2, xl, nullptr, NTN, 64, 128, 0, stream);
  launch_gemm(x1, Wr2, xr, nullptr, NTN, 64, 128, 0, stream);
  fill_f32<<<cdiv((long)NTN * 4, B), B, 0, stream>>>(mx, -INFINITY, (long)NTN * 4);
  fill_f32<<<cdiv((long)NTN * 4, B), B, 0, stream>>>(dn, 0.f, (long)NTN * 4);
  fill_f32<<<cdiv((long)NTN * 128, B), B, 0, stream>>>(accg, 0.f, (long)NTN * 128);
  gat_logits<<<cdiv((long)Etot * 32, B), B, 0, stream>>>(xl, xr, gsrc, gdst, att2, lg, mx, EGN, Etot, 32);
  gat_exp<<<cdiv((long)Etot * 4, B), B, 0, stream>>>(gdst, lg, mx, dn, EGN, Etot);
  gat_scatter<<<cdiv((long)Etot * 32, B), B, 0, stream>>>(xl, gsrc, gdst, lg, dn, accg, EGN, Etot, 32, 5);
  gat_finalize<<<cdiv((long)NTN * 32, B), B, 0, stream>>>(accg, bias2, x2, NTN, 32, 0);

  // ================= JK concat + final linear =================
  concat_jk<<<cdiv((long)NTN * 224, B), B, 0, stream>>>(x0, x1, x2, xjk, NTN);
  launch_gemm(xjk, W_jk, (float*)d_out, b_jk, NTN, 224, 128, 0, stream);
}